// GIN_89627377533173
// MI455X (gfx1250) — compile-verified
//
#include <hip/hip_runtime.h>

#define EMB 300
#define HID 600
#define EMBP 304     // EMB padded to 16
#define HIDP 608     // HID padded to 16
#define NLAYERS 5
#define NGRAPH 512
#define BN_EPS 1e-5f

typedef float v2f __attribute__((ext_vector_type(2)));
typedef float v8f __attribute__((ext_vector_type(8)));

__device__ __forceinline__ void atomAddF(float* p, float v) {
    (void)__hip_atomic_fetch_add(p, v, __ATOMIC_RELAXED, __HIP_MEMORY_SCOPE_AGENT);
}

// ---------------- utility kernels ----------------
__global__ void k_copy(const float* __restrict__ s, float* __restrict__ d, long n) {
    long i = (long)blockIdx.x * blockDim.x + threadIdx.x;
    if (i < n) d[i] = s[i];
}
__global__ void k_zero(float* __restrict__ p, long n) {
    long i = (long)blockIdx.x * blockDim.x + threadIdx.x;
    if (i < n) p[i] = 0.0f;
}

// ---------------- edge message + scatter-add ----------------
__global__ __launch_bounds__(320) void k_edge(const float* __restrict__ h,
                                              const float* __restrict__ ee1,
                                              const float* __restrict__ ee2,
                                              const int* __restrict__ srcI,
                                              const int* __restrict__ dstI,
                                              const int* __restrict__ eattr,
                                              float* __restrict__ agg) {
    int e = blockIdx.x;
    int f = threadIdx.x;
    if (f >= EMB) return;
    int s  = srcI[e];
    int d  = dstI[e];
    int a0 = eattr[2 * e + 0];
    int a1 = eattr[2 * e + 1];
    float v = h[(long)s * EMB + f] + ee1[a0 * EMB + f] + ee2[a1 * EMB + f];
    atomAddF(&agg[(long)d * EMB + f], v);
}

// ---------------- fused MLP (fp32 WMMA, 2 M-tiles per wave) ----------------
// One wave32 per block; block owns 32 node rows (two 16-row WMMA tiles that
// share every B (weight) fragment -> halves weight traffic per output row).
//   zin = agg + h               (32 x 300, staged in LDS)
//   t   = relu(zin @ w1 + b1)   (32 x 600, staged in LDS)
//   z   = t @ w2 + b2           (written in-place into agg)
// Also accumulates per-column sum / sum-of-squares for BatchNorm.
//
// Pad columns (600->608, 300->304) clamp the weight-column index; pad lanes
// compute finite garbage that is never stored (global stores & BN stats are
// guarded in the epilogue; LDS pad columns of `t` are never consumed as K
// inputs since the stage-2 K loop stops at 599). Inner loops are branchless.
__global__ __launch_bounds__(32) void k_mlp(const float* __restrict__ h,
                                            float* __restrict__ zio,     // agg in, z out
                                            const float* __restrict__ w1,
                                            const float* __restrict__ b1,
                                            const float* __restrict__ w2,
                                            const float* __restrict__ b2,
                                            float* __restrict__ stat_sum,
                                            float* __restrict__ stat_sq,
                                            int nrows) {
    __shared__ float zin[32 * EMBP];   // 38912 B
    __shared__ float t[32 * HIDP];     // 77824 B   (total 114 KB < 320 KB/WGP)

    const int lane = threadIdx.x;          // 0..31
    const int l16  = lane & 15;
    const int half = lane >> 4;            // 0: K 0/1, M r ; 1: K 2/3, M r+8
    const int koff = 2 * half;
    const int row0 = blockIdx.x * 32;

    // stage zin = agg + h into LDS (zero padding)
    for (int r = 0; r < 32; ++r) {
        int gr = row0 + r;
        for (int c = lane; c < EMBP; c += 32) {
            float v = 0.0f;
            if (gr < nrows && c < EMB)
                v = zio[(long)gr * EMB + c] + h[(long)gr * EMB + c];
            zin[r * EMBP + c] = v;
        }
    }
    __syncthreads();

    // ---- stage 1: t = relu(zin @ w1 + b1), 38 N-tiles, K = 300 ----
    for (int n = 0; n < HIDP / 16; ++n) {
        const int col  = n * 16 + l16;
        const int bcol = col < HID ? col : (HID - 1);        // clamp, hoisted
        const float* __restrict__ wc = w1 + bcol;
        v8f c0 = {};
        v8f c1 = {};
        for (int k = 0; k < EMB; k += 4) {
            v2f b;
            b.x = wc[(k + koff) * HID];
            b.y = wc[(k + koff + 1) * HID];
            v2f a0 = *(const v2f*)&zin[l16 * EMBP + k + koff];
            v2f a1 = *(const v2f*)&zin[(16 + l16) * EMBP + k + koff];
            c0 = __builtin_amdgcn_wmma_f32_16x16x4_f32(false, a0, false, b,
                                                       (short)0, c0, false, false);
            c1 = __builtin_amdgcn_wmma_f32_16x16x4_f32(false, a1, false, b,
                                                       (short)0, c1, false, false);
        }
        float bias = b1[bcol];
        #pragma unroll
        for (int r = 0; r < 8; ++r) {
            int m = half * 8 + r;
            t[m * HIDP + n * 16 + l16]        = fmaxf(c0[r] + bias, 0.0f);
            t[(16 + m) * HIDP + n * 16 + l16] = fmaxf(c1[r] + bias, 0.0f);
        }
    }
    __syncthreads();

    // ---- stage 2: z = t @ w2 + b2, 19 N-tiles, K = 600; fused BN stats ----
    for (int n = 0; n < EMBP / 16; ++n) {
        const int col  = n * 16 + l16;
        const int bcol = col < EMB ? col : (EMB - 1);        // clamp, hoisted
        const float* __restrict__ wc = w2 + bcol;
        v8f c0 = {};
        v8f c1 = {};
        for (int k = 0; k < HID; k += 4) {
            v2f b;
            b.x = wc[(k + koff) * EMB];
            b.y = wc[(k + koff + 1) * EMB];
            v2f a0 = *(const v2f*)&t[l16 * HIDP + k + koff];
            v2f a1 = *(const v2f*)&t[(16 + l16) * HIDP + k + koff];
            c0 = __builtin_amdgcn_wmma_f32_16x16x4_f32(false, a0, false, b,
                                                       (short)0, c0, false, false);
            c1 = __builtin_amdgcn_wmma_f32_16x16x4_f32(false, a1, false, b,
                                                       (short)0, c1, false, false);
        }
        const bool colOK = (col < EMB);
        float bias = b2[bcol];
        float csum = 0.0f, csq = 0.0f;
        #pragma unroll
        for (int r = 0; r < 8; ++r) {
            int m   = half * 8 + r;
            int gr0 = row0 + m;
            int gr1 = row0 + 16 + m;
            float v0 = c0[r] + bias;
            float v1 = c1[r] + bias;
            if (gr0 < nrows && colOK) zio[(long)gr0 * EMB + col] = v0;
            if (gr1 < nrows && colOK) zio[(long)gr1 * EMB + col] = v1;
            if (gr0 >= nrows || !colOK) v0 = 0.0f;           // exclude pads from stats
            if (gr1 >= nrows || !colOK) v1 = 0.0f;
            csum += v0 + v1;
            csq  += v0 * v0 + v1 * v1;
        }
        // combine the two half-wave contributions for the same column
        csum += __shfl_xor(csum, 16, 32);
        csq  += __shfl_xor(csq, 16, 32);
        if (half == 0 && colOK) {
            atomAddF(&stat_sum[col], csum);
            atomAddF(&stat_sq[col], csq);
        }
    }
}

// ---------------- BatchNorm finalize: scale/shift per column ----------------
__global__ __launch_bounds__(320) void k_bnfin(const float* __restrict__ sum,
                                               const float* __restrict__ sq,
                                               const float* __restrict__ gamma,
                                               const float* __restrict__ beta,
                                               float* __restrict__ scale,
                                               float* __restrict__ shift,
                                               float invN) {
    int c = threadIdx.x;
    if (c >= EMB) return;
    float mean = sum[c] * invN;
    float var  = sq[c] * invN - mean * mean;
    float s = rsqrtf(var + BN_EPS) * gamma[c];
    scale[c] = s;
    shift[c] = beta[c] - mean * s;
}

// ---------------- BN apply + relu -> h ----------------
__global__ void k_bnrelu(const float* __restrict__ z,
                         const float* __restrict__ scale,
                         const float* __restrict__ shift,
                         float* __restrict__ h, long n) {
    long i = (long)blockIdx.x * blockDim.x + threadIdx.x;
    if (i >= n) return;
    int c = (int)(i % EMB);
    float v = z[i] * scale[c] + shift[c];
    h[i] = fmaxf(v, 0.0f);
}

// ---------------- graph pooling ----------------
__global__ __launch_bounds__(320) void k_pool(const float* __restrict__ h,
                                              const int* __restrict__ batch,
                                              float* __restrict__ gsum,
                                              float* __restrict__ gcnt) {
    int node = blockIdx.x;
    int f = threadIdx.x;
    if (f >= EMB) return;
    int b = batch[node];
    atomAddF(&gsum[(long)b * EMB + f], h[(long)node * EMB + f]);
    if (f == 0) atomAddF(&gcnt[b], 1.0f);
}
__global__ void k_pooldiv(float* __restrict__ gsum, const float* __restrict__ gcnt, long n) {
    long i = (long)blockIdx.x * blockDim.x + threadIdx.x;
    if (i >= n) return;
    float c = gcnt[i / EMB];
    gsum[i] = gsum[i] / fmaxf(c, 1.0f);
}

// ---------------- host launch ----------------
extern "C" void kernel_launch(void* const* d_in, const int* in_sizes, int n_in,
                              void* d_out, int out_size, void* d_ws, size_t ws_size,
                              hipStream_t stream) {
    const float* x     = (const float*)d_in[0];
    const float* ee1   = (const float*)d_in[1];   // [5][6][300]
    const float* ee2   = (const float*)d_in[2];   // [5][3][300]
    const float* w1    = (const float*)d_in[3];   // [5][300][600]
    const float* b1    = (const float*)d_in[4];   // [5][600]
    const float* w2    = (const float*)d_in[5];   // [5][600][300]
    const float* b2    = (const float*)d_in[6];   // [5][300]
    const float* gamma = (const float*)d_in[7];   // [5][300]
    const float* beta  = (const float*)d_in[8];   // [5][300]
    const int*   eidx  = (const int*)d_in[9];     // [2][E]
    const int*   eattr = (const int*)d_in[10];    // [E][2]
    const int*   batch = (const int*)d_in[11];    // [N]

    const int N = in_sizes[0] / EMB;
    const int E = in_sizes[10] / 2;
    const long tot = (long)N * EMB;

    // node embeddings live directly in d_out; graph embeddings follow them
    float* h    = (float*)d_out;
    float* gsum = h + tot;

    // workspace: agg (N*300) | stats (4*EMBP) | graph counts (512)
    float* agg  = (float*)d_ws;
    float* stat = agg + tot;                 // sum | sq | scale | shift
    float* gcnt = stat + 4 * EMBP;

    const int TB = 256;
    dim3 gTot((unsigned)((tot + TB - 1) / TB));

    // h = x
    k_copy<<<gTot, TB, 0, stream>>>(x, h, tot);

    const int rowTiles = (N + 31) / 32;
    for (int L = 0; L < NLAYERS; ++L) {
        k_zero<<<gTot, TB, 0, stream>>>(agg, tot);
        k_zero<<<1, 4 * EMBP, 0, stream>>>(stat, 4 * EMBP);

        k_edge<<<E, 320, 0, stream>>>(h,
                                      ee1 + (long)L * 6 * EMB,
                                      ee2 + (long)L * 3 * EMB,
                                      eidx, eidx + E, eattr, agg);

        k_mlp<<<rowTiles, 32, 0, stream>>>(h, agg,
                                           w1 + (long)L * EMB * HID,
                                           b1 + (long)L * HID,
                                           w2 + (long)L * HID * EMB,
                                           b2 + (long)L * EMB,
                                           stat, stat + EMBP, N);

        k_bnfin<<<1, 320, 0, stream>>>(stat, stat + EMBP,
                                       gamma + (long)L * EMB,
                                       beta + (long)L * EMB,
                                       stat + 2 * EMBP, stat + 3 * EMBP,
                                       1.0f / (float)N);

        k_bnrelu<<<gTot, TB, 0, stream>>>(agg, stat + 2 * EMBP, stat + 3 * EMBP, h, tot);
    }

    // graph pooling
    const long gtot = (long)NGRAPH * EMB;
    k_zero<<<(unsigned)((gtot + TB - 1) / TB), TB, 0, stream>>>(gsum, gtot);
    k_zero<<<1, NGRAPH, 0, stream>>>(gcnt, NGRAPH);
    k_pool<<<N, 320, 0, stream>>>(h, batch, gsum, gcnt);
    k_pooldiv<<<(unsigned)((gtot + TB - 1) / TB), TB, 0, stream>>>(gsum, gcnt, gtot);
}